// HR2HK_Gamma_Only_20572893348010
// MI455X (gfx1250) — compile-verified
//
#include <hip/hip_runtime.h>

typedef unsigned int v4u __attribute__((ext_vector_type(4)));
typedef int          v4i __attribute__((ext_vector_type(4)));
typedef int          v8i __attribute__((ext_vector_type(8)));

#define NTOT 10500            // total orbitals: 500*(8+13)
#define FEAT 107

// TDM staging tile: 32 KB of zeros in LDS, streamed out 32 KB per descriptor.
#define TILE_BYTES   32768
#define TILE_UNITS   4096     // 8-byte units (data_size code 3)
#define TILES_PER_WG 8

#if defined(__has_builtin)
#  if __has_builtin(__builtin_amdgcn_tensor_store_from_lds)
#    define HAVE_TDM 1
#  endif
#endif
#ifndef HAVE_TDM
#  define HAVE_TDM 0
#endif

// Upper-shell-triangle enumeration of the 13x13 block (shells 1s 2s 1p 2p 1d).
// Feature k maps to block position (c_OI[k], c_OJ[k]) with factor c_FK[k].
__device__ __constant__ unsigned char c_OI[FEAT] = {
  0,0,0,0,0,0,0,0,0,0,0,0,0,
  1,1,1,1,1,1,1,1,1,1,1,1,
  2,2,2,3,3,3,4,4,4,
  2,2,2,3,3,3,4,4,4,
  2,2,2,2,2,3,3,3,3,3,4,4,4,4,4,
  5,5,5,6,6,6,7,7,7,
  5,5,5,5,5,6,6,6,6,6,7,7,7,7,7,
  8,8,8,8,8,9,9,9,9,9,10,10,10,10,10,
  11,11,11,11,11,12,12,12,12,12
};
__device__ __constant__ unsigned char c_OJ[FEAT] = {
  0,1,2,3,4,5,6,7,8,9,10,11,12,
  1,2,3,4,5,6,7,8,9,10,11,12,
  2,3,4,2,3,4,2,3,4,
  5,6,7,5,6,7,5,6,7,
  8,9,10,11,12,8,9,10,11,12,8,9,10,11,12,
  5,6,7,5,6,7,5,6,7,
  8,9,10,11,12,8,9,10,11,12,8,9,10,11,12,
  8,9,10,11,12,8,9,10,11,12,8,9,10,11,12,
  8,9,10,11,12,8,9,10,11,12
};
__device__ __constant__ float c_FK[FEAT] = {
  0.5f, 1,1,1,1,1,1,1,1,1,1,1,1,
  0.5f, 1,1,1,1,1,1,1,1,1,1,1,
  0.5f,0.5f,0.5f,0.5f,0.5f,0.5f,0.5f,0.5f,0.5f,
  1,1,1,1,1,1,1,1,1,
  1,1,1,1,1,1,1,1,1,1,1,1,1,1,1,
  0.5f,0.5f,0.5f,0.5f,0.5f,0.5f,0.5f,0.5f,0.5f,
  1,1,1,1,1,1,1,1,1,1,1,1,1,1,1,
  0.5f,0.5f,0.5f,0.5f,0.5f,0.5f,0.5f,0.5f,0.5f,0.5f,
  0.5f,0.5f,0.5f,0.5f,0.5f,0.5f,0.5f,0.5f,0.5f,0.5f,
  0.5f,0.5f,0.5f,0.5f,0.5f
};

__device__ __forceinline__ int atom_off(int a) { return (a >> 1) * 21 + (a & 1) * 8; }

#if HAVE_TDM
// ---------------------------------------------------------------------------
// Kernel 1 (TDM path): zero-fill 441 MB output with the Tensor Data Mover.
// Each workgroup zeroes a 32 KB LDS tile once, then one wave enqueues up to 8
// tensor_store_from_lds descriptors (1-D tile, 8-byte elements, 4096 units =
// 32 KB per DMA op), tracked by TENSORcnt and drained with s_wait_tensorcnt.
// ---------------------------------------------------------------------------
__global__ __launch_bounds__(256)
void zero_tdm(float* __restrict__ out, unsigned long long total_units, int total_tiles) {
  __shared__ __align__(16) float zbuf[TILE_BYTES / 4];
  float4* z4 = (float4*)zbuf;
  for (int t = threadIdx.x; t < TILE_BYTES / 16; t += blockDim.x)
    z4[t] = make_float4(0.f, 0.f, 0.f, 0.f);
  __syncthreads();

  if (threadIdx.x < 32) {              // wave 0 issues the DMA descriptors
    unsigned int lds = (unsigned int)(unsigned long long)(&zbuf[0]); // LDS byte offset
    int t0 = blockIdx.x * TILES_PER_WG;
    int t1 = t0 + TILES_PER_WG;
    if (t1 > total_tiles) t1 = total_tiles;
    for (int t = t0; t < t1; ++t) {
      unsigned long long ustart = (unsigned long long)t * TILE_UNITS;
      unsigned long long rem    = total_units - ustart;
      unsigned int units = (unsigned int)(rem < TILE_UNITS ? rem : (unsigned long long)TILE_UNITS);
      unsigned long long ga = (unsigned long long)out + (ustart << 3);

      // D# group 0: count=1 | lds_addr | global_addr[56:0] | type=2
      v4u g0 = { 1u,
                 lds,
                 (unsigned int)ga,
                 (unsigned int)(ga >> 32) | (2u << 30) };
      // D# group 1: workgroup_mask=0, data_size=3 (8B), no barrier/iterate/pad;
      // tensor_dim0 = tile_dim0 = units, tensor_dim1 = 1, dim0_stride = units.
      v8i g1 = { (int)(3u << 16),
                 (int)((units & 0xFFFFu) << 16),          // tensor_dim0[15:0]
                 (int)((units >> 16) | (1u << 16)),       // tensor_dim0[31:16] | tensor_dim1[15:0]=1
                 (int)(units << 16),                      // tensor_dim1[31:16]=0 | tile_dim0
                 0,                                       // tile_dim1=0, tile_dim2=0 (unused)
                 (int)units,                              // tensor_dim0_stride[31:0]
                 0, 0 };
      v4i g2 = { 0, 0, 0, 0 };
      v4i g3 = { 0, 0, 0, 0 };
#if __clang_major__ >= 23
      v8i g4 = { 0, 0, 0, 0, 0, 0, 0, 0 };
      __builtin_amdgcn_tensor_store_from_lds(g0, g1, g2, g3, g4, 0);
#else
      __builtin_amdgcn_tensor_store_from_lds(g0, g1, g2, g3, 0);
#endif
    }
    __builtin_amdgcn_s_wait_tensorcnt(0);
  }
}
#endif // HAVE_TDM

// ---------------------------------------------------------------------------
// Kernel 1 (fallback): async LDS->global streaming stores (ASYNCcnt-tracked).
// ---------------------------------------------------------------------------
__global__ __launch_bounds__(256)
void zero_async(float* __restrict__ out, unsigned long long n16) {
  __shared__ float zbuf[64];
  for (int t = threadIdx.x; t < 64; t += blockDim.x) zbuf[t] = 0.0f;
  __syncthreads();

  unsigned int lds_off = (unsigned int)(unsigned long long)(&zbuf[0]);
  unsigned long long base = (unsigned long long)out;

  unsigned long long idx = (unsigned long long)blockIdx.x * blockDim.x + threadIdx.x;
  unsigned long long stride = (unsigned long long)gridDim.x * blockDim.x;
  for (; idx < n16; idx += stride) {
    unsigned int goff = (unsigned int)(idx << 4);
    asm volatile("global_store_async_from_lds_b128 %0, %1, %2"
                 :: "v"(goff), "v"(lds_off), "s"(base) : "memory");
  }
  asm volatile("s_wait_asynccnt 0x0" ::: "memory");
}

// ---------------------------------------------------------------------------
// Kernel 2: one wave32 per edge. Coalesced read of the 107 features, masked
// expansion, double atomic scatter (fuses the hermitian completion: each
// contribution lands at (r,c) and (c,r), which is exactly block + block.T).
// ---------------------------------------------------------------------------
__global__ __launch_bounds__(256)
void scatter_edges(const float* __restrict__ hop,
                   const int* __restrict__ ei, const int* __restrict__ ej,
                   const int* __restrict__ atype,
                   float* __restrict__ out, int nE) {
  int w = (int)((blockIdx.x * (unsigned)blockDim.x + threadIdx.x) >> 5);
  int lane = threadIdx.x & 31;
  if (w >= nE) return;

  int ai = ei[w], aj = ej[w];
  int ni = 8 + 5 * atype[ai];
  int nj = 8 + 5 * atype[aj];
  int r0 = atom_off(ai);
  int c0 = atom_off(aj);
  const float* f = hop + (size_t)w * FEAT;

  for (int k = lane; k < FEAT; k += 32) {
    int oi = c_OI[k], oj = c_OJ[k];
    if (oi < ni && oj < nj) {
      float v = f[k] * c_FK[k];
      size_t r = (size_t)(r0 + oi), c = (size_t)(c0 + oj);
      atomicAdd(out + r * NTOT + c, v);
      atomicAdd(out + c * NTOT + r, v);
    }
  }
}

// ---------------------------------------------------------------------------
// Kernel 3: one wave32 per atom, onsite diagonal blocks (same double scatter).
// ---------------------------------------------------------------------------
__global__ __launch_bounds__(256)
void scatter_onsite(const float* __restrict__ ons,
                    const int* __restrict__ atype,
                    float* __restrict__ out, int nA) {
  int a = (int)((blockIdx.x * (unsigned)blockDim.x + threadIdx.x) >> 5);
  int lane = threadIdx.x & 31;
  if (a >= nA) return;

  int n = 8 + 5 * atype[a];
  int r0 = atom_off(a);
  const float* f = ons + (size_t)a * FEAT;

  for (int k = lane; k < FEAT; k += 32) {
    int oi = c_OI[k], oj = c_OJ[k];
    if (oi < n && oj < n) {
      float v = f[k] * c_FK[k];
      size_t r = (size_t)(r0 + oi), c = (size_t)(r0 + oj);
      atomicAdd(out + r * NTOT + c, v);
      atomicAdd(out + c * NTOT + r, v);
    }
  }
}

extern "C" void kernel_launch(void* const* d_in, const int* in_sizes, int n_in,
                              void* d_out, int out_size, void* d_ws, size_t ws_size,
                              hipStream_t stream) {
  const float* hop   = (const float*)d_in[0];  // [30000, 107]
  const float* ons   = (const float*)d_in[1];  // [1000, 107]
  const int*   atype = (const int*)d_in[2];    // [1000]
  const int*   eidx  = (const int*)d_in[3];    // [2, 30000] flat
  float* out = (float*)d_out;                  // [10500, 10500]

  int nE = in_sizes[3] / 2;
  int nA = in_sizes[2];

#if HAVE_TDM
  unsigned long long units = (unsigned long long)out_size / 2;   // 8-byte units
  int tiles  = (int)((units + TILE_UNITS - 1) / TILE_UNITS);
  int blocks = (tiles + TILES_PER_WG - 1) / TILES_PER_WG;
  zero_tdm<<<blocks, 256, 0, stream>>>(out, units, tiles);
#else
  unsigned long long n16 = (unsigned long long)out_size / 4;     // 16-byte chunks
  zero_async<<<4096, 256, 0, stream>>>(out, n16);
#endif

  scatter_edges<<<(nE + 7) / 8, 256, 0, stream>>>(hop, eidx, eidx + nE, atype, out, nE);
  scatter_onsite<<<(nA + 7) / 8, 256, 0, stream>>>(ons, atype, out, nA);
}